// ECELoss_26414048870985
// MI455X (gfx1250) — compile-verified
//
#include <hip/hip_runtime.h>
#include <hip/hip_bf16.h>
#include <stdint.h>

#define NBINS            15
#define NCLASSES         128
#define WAVES_PER_BLOCK  8
#define BLOCK_THREADS    256
#define GRID_BLOCKS      2048
#define DEPTH            4          // async pipeline stages per wave

// ---------------------------------------------------------------------------
// CDNA5 async data-mover helpers (gfx1250): global -> LDS bulk copy tracked by
// ASYNCcnt.  One instruction moves 32 lanes x 16B = 512B (a full row).
// TH_LOAD_NT: probs is a 1GB read-once stream (> 192MB L2), evict-first.
// ---------------------------------------------------------------------------
__device__ __forceinline__ void async_row_copy16(const float* gsrc, const float* ldst) {
    // Low 32 bits of a flat shared pointer are the LDS byte offset
    // (aperture layout, CDNA5 ISA 10.2): use directly as the VDST lds addr.
    unsigned lds = (unsigned)(uintptr_t)ldst;
    asm volatile("global_load_async_to_lds_b128 %0, %1, off th:TH_LOAD_NT"
                 :
                 : "v"(lds), "v"(gsrc)
                 : "memory");
}
__device__ __forceinline__ void wait_async_le3() { asm volatile("s_wait_asynccnt 0x3" ::: "memory"); }
__device__ __forceinline__ void wait_async_le2() { asm volatile("s_wait_asynccnt 0x2" ::: "memory"); }
__device__ __forceinline__ void wait_async_le1() { asm volatile("s_wait_asynccnt 0x1" ::: "memory"); }
__device__ __forceinline__ void wait_async_le0() { asm volatile("s_wait_asynccnt 0x0" ::: "memory"); }
__device__ __forceinline__ void wait_ds0()       { asm volatile("s_wait_dscnt 0x0"    ::: "memory"); }

// ---------------------------------------------------------------------------
// Per-row reduction + binning shared by steady/tail phases.
// ---------------------------------------------------------------------------
__device__ __forceinline__ void process_row(const float4 v, int lane, int is_logit,
                                            int lab, float* sbins) {
    // per-lane max / argmax over 4 columns (base col = lane*4)
    float m  = v.x;
    int   mi = lane * 4;
    if (v.y > m) { m = v.y; mi = lane * 4 + 1; }
    if (v.z > m) { m = v.z; mi = lane * 4 + 2; }
    if (v.w > m) { m = v.w; mi = lane * 4 + 3; }

    // wave32 butterfly reduce: max value, lowest index on tie
    #pragma unroll
    for (int s = 16; s >= 1; s >>= 1) {
        float om  = __shfl_xor(m, s, 32);
        int   omi = __shfl_xor(mi, s, 32);
        if (om > m || (om == m && omi < mi)) { m = om; mi = omi; }
    }

    float conf;
    if (is_logit) {
        // max(softmax(x)) = 1 / sum_j exp(x_j - max)
        float e = __expf(v.x - m) + __expf(v.y - m) +
                  __expf(v.z - m) + __expf(v.w - m);
        #pragma unroll
        for (int s = 16; s >= 1; s >>= 1) e += __shfl_xor(e, s, 32);
        conf = 1.0f / e;
    } else {
        conf = m;
    }

    if (lane == 0) {
        int bin = (int)ceilf(conf * (float)NBINS) - 1;
        bin = bin < 0 ? 0 : (bin > NBINS - 1 ? NBINS - 1 : bin);
        const float acc = (mi == lab) ? 1.0f : 0.0f;
        atomicAdd(&sbins[bin],             1.0f);
        atomicAdd(&sbins[NBINS + bin],     conf);
        atomicAdd(&sbins[2 * NBINS + bin], acc);
    }
}

// ---------------------------------------------------------------------------
// Main kernel: one wave32 per row.  4-deep async global->LDS pipeline,
// wave-level max/argmax + exp-sum reductions, per-block LDS bin accumulators.
// ---------------------------------------------------------------------------
__global__ __launch_bounds__(BLOCK_THREADS)
void ece_bins_kernel(const float* __restrict__ probs,
                     const int*   __restrict__ labels,
                     const int*   __restrict__ is_logit_p,
                     float*       __restrict__ ws,
                     int n_rows) {
    __shared__ __align__(16) float srow[WAVES_PER_BLOCK * DEPTH * NCLASSES];
    __shared__ float sbins[3 * NBINS];

    const int tid = threadIdx.x;
    if (tid < 3 * NBINS) sbins[tid] = 0.0f;
    __syncthreads();

    const int lane     = tid & 31;
    const int wid      = tid >> 5;
    const int is_logit = *is_logit_p;

    float* buf = &srow[wid * DEPTH * NCLASSES];

    const long long r0     = (long long)blockIdx.x * WAVES_PER_BLOCK + wid;
    const long long stride = (long long)gridDim.x * WAVES_PER_BLOCK;

    // iterations (rows) this wave owns
    long long rem = (r0 < n_rows) ? ((n_rows - 1 - r0) / stride + 1) : 0;

    // prologue: fill up to DEPTH pipeline stages
    #pragma unroll
    for (int j = 0; j < DEPTH; ++j) {
        const long long rj = r0 + (long long)j * stride;
        if (rj < n_rows) {
            async_row_copy16(probs + rj * NCLASSES + lane * 4,
                             buf + j * NCLASSES + lane * 4);
        }
    }

    long long r = r0;
    int k = 0;

    // steady phase: always DEPTH copies outstanding, one refill per iteration
    while (rem > DEPTH) {
        wait_async_le3();                          // oldest of 4 (row r) landed
        const float4 v = *reinterpret_cast<const float4*>(buf + k * NCLASSES + lane * 4);
        const int lab  = labels[r];                // overlap latency with compute
        wait_ds0();                                // reads in VGPRs -> safe to overwrite
        async_row_copy16(probs + (r + (long long)DEPTH * stride) * NCLASSES + lane * 4,
                         buf + k * NCLASSES + lane * 4);

        process_row(v, lane, is_logit, lab, sbins);

        r += stride; --rem; k = (k + 1) & (DEPTH - 1);
    }

    // tail phase: <= DEPTH rows left, outstanding == rem, no more refills
    while (r < n_rows) {
        if      (rem >= 4) wait_async_le3();
        else if (rem == 3) wait_async_le2();
        else if (rem == 2) wait_async_le1();
        else               wait_async_le0();
        const float4 v = *reinterpret_cast<const float4*>(buf + k * NCLASSES + lane * 4);
        const int lab  = labels[r];

        process_row(v, lane, is_logit, lab, sbins);

        r += stride; --rem; k = (k + 1) & (DEPTH - 1);
    }

    __syncthreads();
    if (tid < 3 * NBINS) atomicAdd(&ws[tid], sbins[tid]);
}

// ---------------------------------------------------------------------------
// Finalize: fold 15 bins into the ECE scalar.
// ---------------------------------------------------------------------------
__global__ void ece_finalize_kernel(const float* __restrict__ ws,
                                    float* __restrict__ out,
                                    int n_rows) {
    if (threadIdx.x == 0 && blockIdx.x == 0) {
        const float n = (float)n_rows;
        float ece = 0.0f;
        for (int b = 0; b < NBINS; ++b) {
            const float c = ws[b];
            if (c > 0.0f) {
                const float gap = fabsf(ws[NBINS + b] / c - ws[2 * NBINS + b] / c);
                ece += gap * (c / n);
            }
        }
        out[0] = ece;
    }
}

extern "C" void kernel_launch(void* const* d_in, const int* in_sizes, int n_in,
                              void* d_out, int out_size, void* d_ws, size_t ws_size,
                              hipStream_t stream) {
    const float* probs    = (const float*)d_in[0];
    const int*   labels   = (const int*)d_in[1];
    const int*   is_logit = (const int*)d_in[2];
    float*       out      = (float*)d_out;
    float*       ws       = (float*)d_ws;
    const int    n_rows   = in_sizes[1];   // labels: one per sample

    (void)n_in; (void)out_size; (void)ws_size;

    hipMemsetAsync(ws, 0, 3 * NBINS * sizeof(float), stream);
    ece_bins_kernel<<<GRID_BLOCKS, BLOCK_THREADS, 0, stream>>>(
        probs, labels, is_logit, ws, n_rows);
    ece_finalize_kernel<<<1, 32, 0, stream>>>(ws, out, n_rows);
}